// LinearGaussianQ_25658134626491
// MI455X (gfx1250) — compile-verified
//
#include <hip/hip_runtime.h>
#include <math.h>

// ---------------- problem constants ----------------
static constexpr int   TSTEPS = 4096;
static constexpr int   ZD     = 64;    // DZ
static constexpr int   XD     = 32;    // DX
static constexpr int   ZD2    = ZD * ZD;   // 4096
static constexpr int   XD2    = XD * XD;   // 1024
static constexpr float F_LOG2PI = 1.8378770664093453f;

// ---------------- workspace layout (float offsets) ----------------
static constexpr size_t O_PTR  = 0;                                  // inv(p_Q)            ZD2
static constexpr size_t O_PER  = O_PTR  + ZD2;                       // inv(p_R)            XD2
static constexpr size_t O_QTR  = O_PER  + XD2;                       // inv(q_Q)            ZD2
static constexpr size_t O_P0I  = O_QTR  + ZD2;                       // inv(p_P0)           ZD2
static constexpr size_t O_CTQ  = O_P0I  + ZD2;                       // q_A^T inv(q_Q)      ZD2
static constexpr size_t O_CTQA = O_CTQ  + ZD2;                       // CTQ q_A             ZD2
static constexpr size_t O_MM   = O_CTQA + ZD2;                       // p_A^T inv(p_Q) p_A  ZD2
static constexpr size_t O_G2   = O_MM   + ZD2;                       // p_C^T O2 p_C        ZD2
static constexpr size_t O_SC   = O_G2   + ZD2;                       // scalars             16
static constexpr size_t O_FM   = O_SC   + 16;                        // filt_m   T*ZD
static constexpr size_t O_FP   = O_FM   + (size_t)TSTEPS * ZD;       // filt_P   T*ZD2
static constexpr size_t O_BA   = O_FP   + (size_t)TSTEPS * ZD2;      // BA       T*ZD2
static constexpr size_t O_BAV  = O_BA   + (size_t)TSTEPS * ZD2;      // Ba       T*ZD
static constexpr size_t O_BC   = O_BAV  + (size_t)TSTEPS * ZD;       // Bcov     T*ZD2
static constexpr size_t O_BLD  = O_BC   + (size_t)TSTEPS * ZD2;      // Blogdet  T
static constexpr size_t O_TRT  = O_BLD  + TSTEPS;                    // tr_term  T
static constexpr size_t O_A1   = O_TRT  + TSTEPS;                    // A1       T*ZD2
static constexpr size_t O_B1   = O_A1   + (size_t)TSTEPS * ZD2;      // b1       T*ZD
static constexpr size_t O_G    = O_B1   + (size_t)TSTEPS * ZD;       // G        T*ZD2
static constexpr size_t O_PF   = O_G    + (size_t)TSTEPS * ZD2;      // Pfull    T*ZD2
static constexpr size_t O_SF   = O_PF   + (size_t)TSTEPS * ZD2;      // sfull    T*ZD

// ---------------- async global -> LDS copy (CDNA5 ASYNCcnt path) ----------------
#if defined(__has_builtin)
#if __has_builtin(__builtin_amdgcn_global_load_async_to_lds_b128) && \
    __has_builtin(__builtin_amdgcn_s_wait_asynccnt)
#define HAVE_ASYNC_LDS 1
#endif
#endif
#ifndef HAVE_ASYNC_LDS
#define HAVE_ASYNC_LDS 0
#endif

typedef int v4i_t __attribute__((vector_size(16)));   // matches builtin param type

// copy one 64x64 f32 matrix (16 KB) global -> LDS; 256 threads x 4 x b128
__device__ __forceinline__ void copy16k_async(const float* g, float* l)
{
#if HAVE_ASYNC_LDS
    const int tid = threadIdx.x;
    for (int r = 0; r < 4; ++r) {
        const int e = r * 1024 + tid * 4;
        __builtin_amdgcn_global_load_async_to_lds_b128((v4i_t*)(g + e),
                                                       (v4i_t*)(l + e), 0, 0);
    }
#else
    const int tid = threadIdx.x;
    for (int i = tid; i < ZD2; i += 256) l[i] = g[i];
#endif
}

__device__ __forceinline__ void wait_async_copies()
{
#if HAVE_ASYNC_LDS
    __builtin_amdgcn_s_wait_asynccnt(0);
#endif
}

// ---------------- FP32 WMMA tile GEMM ----------------
typedef __attribute__((ext_vector_type(2))) float v2f;
typedef __attribute__((ext_vector_type(8))) float v8f;
union V8 { v8f v; float f[8]; };

// D(MxN) = op(A)(MxK) * op(B)(KxN)   (op = optional transpose)
// Pointers generic (LDS or global). Requires blockDim.x == 256 (8 waves, wave32).
__device__ __forceinline__ void mm_wmma(const float* __restrict__ A, int lda, int tA,
                                        const float* __restrict__ B, int ldb, int tB,
                                        float* __restrict__ D, int ldd,
                                        int M, int N, int K)
{
    __syncthreads();
    const int tid  = threadIdx.x;
    const int wave = tid >> 5;
    const int lane = tid & 31;
    const int lr   = lane & 15;   // 0..15
    const int hi   = lane >> 4;   // 0 or 1
    const int tilesN = N >> 4;
    const int ntiles = (M >> 4) * tilesN;
    for (int tile = wave; tile < ntiles; tile += 8) {
        const int tm = (tile / tilesN) << 4;
        const int tn = (tile % tilesN) << 4;
        V8 c;
        for (int v = 0; v < 8; ++v) c.f[v] = 0.0f;
        for (int kb = 0; kb < K; kb += 4) {
            const int am = tm + lr;
            const int ak = kb + hi * 2;
            v2f a, b;
            if (tA) { a.x = A[(size_t)ak * lda + am]; a.y = A[(size_t)(ak + 1) * lda + am]; }
            else    { a.x = A[(size_t)am * lda + ak]; a.y = A[(size_t)am * lda + ak + 1]; }
            const int bn = tn + lr;
            const int bk = kb + hi * 2;
            if (tB) { b.x = B[(size_t)bn * ldb + bk]; b.y = B[(size_t)bn * ldb + bk + 1]; }
            else    { b.x = B[(size_t)bk * ldb + bn]; b.y = B[(size_t)(bk + 1) * ldb + bn]; }
            c.v = __builtin_amdgcn_wmma_f32_16x16x4_f32(false, a, false, b,
                                                        (short)0, c.v, false, false);
        }
        for (int v = 0; v < 8; ++v)
            D[(size_t)(tm + v + hi * 8) * ldd + tn + lr] = c.f[v];
    }
    __syncthreads();
}

// ---------------- Gauss-Jordan inverse (block cooperative) ----------------
__device__ void gauss_jordan(float* aug, int n, float* ld_out)
{
    __shared__ float gj_f[ZD];
    __shared__ float gj_piv;
    const int tid = threadIdx.x;
    const int w = 2 * n;
    for (int col = 0; col < n; ++col) {
        if (tid == 0) {
            float p = aug[col * w + col];
            gj_piv = p;
            *ld_out += logf(fabsf(p));
        }
        __syncthreads();
        const float inv = 1.0f / gj_piv;
        for (int j = tid; j < w; j += 256) aug[col * w + j] *= inv;
        __syncthreads();
        if (tid < n) gj_f[tid] = (tid == col) ? 0.0f : aug[tid * w + col];
        __syncthreads();
        for (int idx = tid; idx < n * w; idx += 256) {
            int i = idx / w, j = idx - i * w;
            if (i != col) aug[i * w + j] -= gj_f[i] * aug[col * w + j];
        }
        __syncthreads();
    }
}

// ================= kernel 1: constant precomputation =================
__global__ __launch_bounds__(256)
void k_precompute(const float* pA, const float* pQ, const float* pC,
                  const float* pR, const float* pP0, const float* qA,
                  const float* qQ, float* ws)
{
    __shared__ float aug[ZD * 2 * ZD];   // 32 KB
    __shared__ float sT[ZD2];            // 16 KB
    __shared__ float ld;
    const int tid = threadIdx.x;

    // ---- inv(p_Q), logdet ----
    for (int i = tid; i < ZD2; i += 256) {
        int r = i >> 6, c = i & 63;
        aug[r * 128 + c] = pQ[i];
        aug[r * 128 + 64 + c] = (r == c) ? 1.f : 0.f;
    }
    if (tid == 0) ld = 0.f;
    __syncthreads();
    gauss_jordan(aug, ZD, &ld);
    for (int i = tid; i < ZD2; i += 256) {
        int r = i >> 6, c = i & 63;
        ws[O_PTR + i] = aug[r * 128 + 64 + c];
    }
    if (tid == 0) ws[O_SC + 0] = ld;
    __syncthreads();

    // ---- inv(q_Q) ----
    for (int i = tid; i < ZD2; i += 256) {
        int r = i >> 6, c = i & 63;
        aug[r * 128 + c] = qQ[i];
        aug[r * 128 + 64 + c] = (r == c) ? 1.f : 0.f;
    }
    if (tid == 0) ld = 0.f;
    __syncthreads();
    gauss_jordan(aug, ZD, &ld);
    for (int i = tid; i < ZD2; i += 256) {
        int r = i >> 6, c = i & 63;
        ws[O_QTR + i] = aug[r * 128 + 64 + c];
    }
    __syncthreads();

    // ---- inv(p_P0), logdet ----
    for (int i = tid; i < ZD2; i += 256) {
        int r = i >> 6, c = i & 63;
        aug[r * 128 + c] = pP0[i];
        aug[r * 128 + 64 + c] = (r == c) ? 1.f : 0.f;
    }
    if (tid == 0) ld = 0.f;
    __syncthreads();
    gauss_jordan(aug, ZD, &ld);
    for (int i = tid; i < ZD2; i += 256) {
        int r = i >> 6, c = i & 63;
        ws[O_P0I + i] = aug[r * 128 + 64 + c];
    }
    if (tid == 0) ws[O_SC + 2] = ld;
    __syncthreads();

    // ---- inv(p_R), logdet ----
    for (int i = tid; i < XD2; i += 256) {
        int r = i >> 5, c = i & 31;
        aug[r * 64 + c] = pR[i];
        aug[r * 64 + 32 + c] = (r == c) ? 1.f : 0.f;
    }
    if (tid == 0) ld = 0.f;
    __syncthreads();
    gauss_jordan(aug, XD, &ld);
    for (int i = tid; i < XD2; i += 256) {
        int r = i >> 5, c = i & 31;
        ws[O_PER + i] = aug[r * 64 + 32 + c];
    }
    if (tid == 0) { ws[O_SC + 1] = ld; ws[O_SC + 3] = 0.f; ws[O_SC + 4] = 0.f; ws[O_SC + 5] = 0.f; }
    __threadfence_block();
    __syncthreads();

    // ---- CTQ = q_A^T inv(q_Q); CTQA = CTQ q_A ----
    mm_wmma(qA, ZD, 1, ws + O_QTR, ZD, 0, ws + O_CTQ, ZD, ZD, ZD, ZD);
    __threadfence_block(); __syncthreads();
    mm_wmma(ws + O_CTQ, ZD, 0, qA, ZD, 0, ws + O_CTQA, ZD, ZD, ZD, ZD);

    // ---- M = p_A^T inv(p_Q) p_A ----
    mm_wmma(ws + O_PTR, ZD, 0, pA, ZD, 0, sT, ZD, ZD, ZD, ZD);
    mm_wmma(pA, ZD, 1, sT, ZD, 0, ws + O_MM, ZD, ZD, ZD, ZD);

    // ---- G2 = p_C^T (-0.5 inv(p_R)) p_C ----
    mm_wmma(ws + O_PER, XD, 0, pC, ZD, 0, aug, ZD, XD, ZD, XD);   // inv(p_R) p_C (32x64)
    for (int i = tid; i < XD * ZD; i += 256) aug[i] *= -0.5f;
    __syncthreads();
    mm_wmma(pC, ZD, 1, aug, ZD, 0, ws + O_G2, ZD, ZD, ZD, XD);
}

// ================= kernel 2: sequential Kalman filter =================
__global__ __launch_bounds__(256)
void k_kalman(const float* obs, const float* qA, const float* qb,
              const float* qQ, const float* qC, const float* qd,
              const float* qR, const float* qmu0, const float* qP0,
              float* ws)
{
    __shared__ float sP[ZD2];             // P          16 KB
    __shared__ float s1[ZD2];             // scratch    16 KB
    __shared__ float s2[ZD2];             // scratch    16 KB
    __shared__ float augS[XD * 2 * XD];   //            8 KB
    __shared__ float cqA[ZD2];            // q_A        16 KB
    __shared__ float cqQ[ZD2];            // q_Q        16 KB
    __shared__ float cqC[XD * ZD];        // q_C        8 KB
    __shared__ float cqR[XD2];            // q_R        4 KB
    __shared__ float m[ZD], mt2[ZD], yv[XD];
    __shared__ float dld;
    const int tid = threadIdx.x;
    float* sCP = s2;            // 32x64 (during update)
    float* sX  = s2 + XD * ZD;  // 32x64

    if (tid < ZD) m[tid] = qmu0[tid];
    for (int i = tid; i < ZD2; i += 256) { sP[i] = qP0[i]; cqA[i] = qA[i]; cqQ[i] = qQ[i]; }
    for (int i = tid; i < XD * ZD; i += 256) cqC[i] = qC[i];
    for (int i = tid; i < XD2; i += 256) cqR[i] = qR[i];
    __syncthreads();

    for (int t = 0; t < TSTEPS; ++t) {
        if (t > 0) {
            // m_pred = q_A m + q_b
            if (tid < ZD) {
                float acc = qb[tid];
                for (int k = 0; k < ZD; ++k) acc += cqA[tid * ZD + k] * m[k];
                mt2[tid] = acc;
            }
            __syncthreads();
            if (tid < ZD) m[tid] = mt2[tid];
            // P_pred = q_A P q_A^T + q_Q
            mm_wmma(cqA, ZD, 0, sP, ZD, 0, s1, ZD, ZD, ZD, ZD);
            mm_wmma(s1, ZD, 0, cqA, ZD, 1, s2, ZD, ZD, ZD, ZD);
            for (int i = tid; i < ZD2; i += 256) sP[i] = s2[i] + cqQ[i];
            __syncthreads();
        }
        // ---- measurement update ----
        mm_wmma(cqC, ZD, 0, sP, ZD, 0, sCP, ZD, XD, ZD, ZD);        // CP = C P (32x64)
        mm_wmma(sCP, ZD, 0, cqC, ZD, 1, s1, XD, XD, XD, ZD);        // S0 = CP C^T (32x32)
        for (int i = tid; i < XD2; i += 256) {
            int r = i >> 5, c = i & 31;
            augS[r * 64 + c] = s1[i] + cqR[i];
            augS[r * 64 + 32 + c] = (r == c) ? 1.f : 0.f;
        }
        if (tid == 0) dld = 0.f;
        __syncthreads();
        gauss_jordan(augS, XD, &dld);                               // S^-1
        mm_wmma(augS + 32, 64, 0, sCP, ZD, 0, sX, ZD, XD, ZD, XD);  // X = S^-1 CP
        // innovation
        if (tid < XD) {
            float acc = obs[(size_t)t * XD + tid] - qd[tid];
            for (int k = 0; k < ZD; ++k) acc -= cqC[tid * ZD + k] * m[k];
            yv[tid] = acc;
        }
        __syncthreads();
        // m += X^T yv   (K = X^T)
        if (tid < ZD) {
            float acc = 0.f;
            for (int j = 0; j < XD; ++j) acc += sX[j * ZD + tid] * yv[j];
            mt2[tid] = m[tid] + acc;
        }
        __syncthreads();
        if (tid < ZD) m[tid] = mt2[tid];
        // IKC = I - X^T q_C
        mm_wmma(sX, ZD, 1, cqC, ZD, 0, s1, ZD, ZD, ZD, XD);
        for (int i = tid; i < ZD2; i += 256) {
            int r = i >> 6, c = i & 63;
            s1[i] = ((r == c) ? 1.f : 0.f) - s1[i];
        }
        __syncthreads();
        mm_wmma(s1, ZD, 0, sP, ZD, 0, s2, ZD, ZD, ZD, ZD);          // P = IKC P
        for (int i = tid; i < ZD2; i += 256) sP[i] = s2[i];
        __syncthreads();
        // store filtering state
        if (tid < ZD) ws[O_FM + (size_t)t * ZD + tid] = m[tid];
        for (int i = tid; i < ZD2; i += 256) ws[O_FP + (size_t)t * ZD2 + i] = sP[i];
        __syncthreads();
    }
}

// ================= kernel 3: backward kernels (parallel over t) =================
__global__ __launch_bounds__(256)
void k_bwd(const float* qb, float* ws)
{
    const int t = blockIdx.x;               // 0..T-2
    __shared__ float aug[ZD * 128];         // 32 KB
    __shared__ float sBc[ZD2];              // 16 KB
    __shared__ float wv[ZD];
    __shared__ float ldacc, red;
    const int tid = threadIdx.x;
    const float* Pt = ws + O_FP + (size_t)t * ZD2;
    const float* mt = ws + O_FM + (size_t)t * ZD;

    for (int i = tid; i < ZD2; i += 256) {
        int r = i >> 6, c = i & 63;
        aug[r * 128 + c] = Pt[i];
        aug[r * 128 + 64 + c] = (r == c) ? 1.f : 0.f;
    }
    if (tid == 0) ldacc = 0.f;
    __syncthreads();
    gauss_jordan(aug, ZD, &ldacc);
    if (tid < ZD) {
        float acc = 0.f;
        for (int k = 0; k < ZD; ++k)
            acc += aug[tid * 128 + 64 + k] * mt[k] - ws[O_CTQ + tid * ZD + k] * qb[k];
        wv[tid] = acc;
    }
    __syncthreads();
    for (int i = tid; i < ZD2; i += 256) {
        int r = i >> 6, c = i & 63;
        aug[r * 128 + c] = ws[O_CTQA + i] + aug[r * 128 + 64 + c];
    }
    __syncthreads();
    for (int i = tid; i < ZD2; i += 256) {
        int r = i >> 6, c = i & 63;
        aug[r * 128 + 64 + c] = (r == c) ? 1.f : 0.f;
    }
    if (tid == 0) ldacc = 0.f;
    __syncthreads();
    gauss_jordan(aug, ZD, &ldacc);
    for (int i = tid; i < ZD2; i += 256) {
        int r = i >> 6, c = i & 63;
        float v = aug[r * 128 + 64 + c];
        sBc[i] = v;
        ws[O_BC + (size_t)t * ZD2 + i] = v;                 // Bcov
    }
    if (tid == 0) ws[O_BLD + t] = -ldacc;                   // logdet(Bcov)
    __syncthreads();
    mm_wmma(sBc, ZD, 0, ws + O_CTQ, ZD, 0, ws + O_BA + (size_t)t * ZD2, ZD, ZD, ZD, ZD);
    if (tid < ZD) {
        float acc = 0.f;
        for (int k = 0; k < ZD; ++k) acc += sBc[tid * ZD + k] * wv[k];
        ws[O_BAV + (size_t)t * ZD + tid] = acc;
    }
    if (tid == 0) red = 0.f;
    __syncthreads();
    float part = 0.f;
    for (int i = tid; i < ZD2; i += 256) {
        int a = i >> 6, b = i & 63;
        part += ws[O_MM + a * ZD + b] * sBc[b * ZD + a];
    }
    atomicAdd(&red, part);
    __syncthreads();
    if (tid == 0) ws[O_TRT + t] = red;
}

// ================= kernel 4: A1/b1/G assembly (parallel over t) =================
__global__ __launch_bounds__(256)
void k_prep(const float* pA, const float* pb, const float* pmu0, float* ws)
{
    const int t = blockIdx.x;
    __shared__ float sA1[ZD2], sT[ZD2], sT2[ZD2];   // 48 KB
    const int tid = threadIdx.x;

    if (t == 0) {
        for (int i = tid; i < ZD2; i += 256) {
            int r = i >> 6, c = i & 63;
            sA1[i] = (r == c) ? 1.f : 0.f;
        }
        if (tid < ZD) ws[O_B1 + tid] = -pmu0[tid];
        __syncthreads();
    } else {
        mm_wmma(pA, ZD, 0, ws + O_BA + (size_t)(t - 1) * ZD2, ZD, 0, sA1, ZD, ZD, ZD, ZD);
        for (int i = tid; i < ZD2; i += 256) {
            int r = i >> 6, c = i & 63;
            sA1[i] -= (r == c) ? 1.f : 0.f;
        }
        if (tid < ZD) {
            float acc = pb[tid];
            const float* Bav = ws + O_BAV + (size_t)(t - 1) * ZD;
            for (int k = 0; k < ZD; ++k) acc += pA[tid * ZD + k] * Bav[k];
            ws[O_B1 + (size_t)t * ZD + tid] = acc;
        }
        __syncthreads();
    }
    for (int i = tid; i < ZD2; i += 256) ws[O_A1 + (size_t)t * ZD2 + i] = sA1[i];
    __syncthreads();

    if (t <= TSTEPS - 2) {
        const float* Prec = (t == 0) ? (ws + O_P0I) : (ws + O_PTR);
        mm_wmma(Prec, ZD, 0, sA1, ZD, 0, sT, ZD, ZD, ZD, ZD);
        mm_wmma(sA1, ZD, 1, sT, ZD, 0, sT2, ZD, ZD, ZD, ZD);
        for (int i = tid; i < ZD2; i += 256)
            ws[O_G + (size_t)t * ZD2 + i] = -0.5f * sT2[i] + ws[O_G2 + i];
    }
}

// ================= kernel 5: both sequential scans, concurrent blocks =================
// block 0: reverse composition scan (Pfull/sfull); block 1: H forward scan.
// BA matrices double-buffered in LDS via async global->LDS copies (ASYNCcnt).
__global__ __launch_bounds__(256)
void k_scan2(float* ws)
{
    __shared__ float pool[5 * ZD2];    // 80 KB
    __shared__ float vec0[ZD], vec1[ZD];
    __shared__ float red, total;
    const int tid = threadIdx.x;

    if (blockIdx.x == 0) {
        // ---------------- reverse composition scan ----------------
        float* sPc  = pool;
        float* sNew = pool + ZD2;
        float* buf0 = pool + 2 * ZD2;
        float* buf1 = pool + 3 * ZD2;
        float* buf[2] = { buf0, buf1 };
        for (int i = tid; i < ZD2; i += 256) {
            int r = i >> 6, c = i & 63;
            float v = (r == c) ? 1.f : 0.f;
            sPc[i] = v;
            ws[O_PF + (size_t)(TSTEPS - 1) * ZD2 + i] = v;
        }
        if (tid < ZD) { vec0[tid] = 0.f; ws[O_SF + (size_t)(TSTEPS - 1) * ZD + tid] = 0.f; }
        copy16k_async(ws + O_BA + (size_t)(TSTEPS - 2) * ZD2, buf[0]);
        wait_async_copies();
        __syncthreads();
        int cur = 0;
        for (int j = TSTEPS - 2; j >= 0; --j) {
            const float* BAl = buf[cur];
            if (j > 0) copy16k_async(ws + O_BA + (size_t)(j - 1) * ZD2, buf[cur ^ 1]);
            mm_wmma(BAl, ZD, 0, sPc, ZD, 0, sNew, ZD, ZD, ZD, ZD);
            if (tid < ZD) {
                float acc = ws[O_BAV + (size_t)j * ZD + tid];
                for (int k = 0; k < ZD; ++k) acc += BAl[tid * ZD + k] * vec0[k];
                vec1[tid] = acc;
            }
            __syncthreads();
            if (tid < ZD) { vec0[tid] = vec1[tid]; ws[O_SF + (size_t)j * ZD + tid] = vec1[tid]; }
            for (int i = tid; i < ZD2; i += 256) {
                sPc[i] = sNew[i];
                ws[O_PF + (size_t)j * ZD2 + i] = sNew[i];
            }
            if (j > 0) wait_async_copies();
            __syncthreads();
            cur ^= 1;
        }
    } else {
        // ---------------- H forward scan ----------------
        float* sH   = pool;
        float* sT   = pool + ZD2;
        float* sT2  = pool + 2 * ZD2;
        float* buf0 = pool + 3 * ZD2;
        float* buf1 = pool + 4 * ZD2;
        float* buf[2] = { buf0, buf1 };
        if (tid == 0) total = 0.f;
        copy16k_async(ws + O_BA, buf[0]);          // BA[0], first used at k = 1
        wait_async_copies();
        __syncthreads();
        for (int k = 0; k <= TSTEPS - 2; ++k) {
            if (k == 0) {
                for (int i = tid; i < ZD2; i += 256) sH[i] = ws[O_G + i];
                __syncthreads();
            } else {
                const float* BAl = buf[(k - 1) & 1];
                if (k < TSTEPS - 2) copy16k_async(ws + O_BA + (size_t)k * ZD2, buf[k & 1]);
                __builtin_prefetch(ws + O_G + (size_t)k * ZD2 + tid * 16, 0, 0);
                mm_wmma(sH, ZD, 0, BAl, ZD, 0, sT, ZD, ZD, ZD, ZD);
                mm_wmma(BAl, ZD, 1, sT, ZD, 0, sT2, ZD, ZD, ZD, ZD);
                for (int i = tid; i < ZD2; i += 256)
                    sH[i] = sT2[i] + ws[O_G + (size_t)k * ZD2 + i];
                if (k < TSTEPS - 2) wait_async_copies();
                __syncthreads();
            }
            if (tid == 0) red = 0.f;
            __syncthreads();
            float part = 0.f;
            const float* Bck = ws + O_BC + (size_t)k * ZD2;
            for (int i = tid; i < ZD2; i += 256) {
                int a = i >> 6, b = i & 63;
                part += sH[a * ZD + b] * Bck[b * ZD + a];
            }
            atomicAdd(&red, part);
            __syncthreads();
            if (tid == 0) total += red;
            __syncthreads();
        }
        if (tid == 0) ws[O_SC + 3] = total;
    }
}

// ================= kernel 6: final integrated quad forms (parallel over t) =================
__global__ __launch_bounds__(256)
void k_final(const float* obs, const float* pC, const float* pd, float* ws)
{
    const int t = blockIdx.x;
    __shared__ float sA[ZD2], sB[ZD2], sD[ZD2];   // 48 KB
    __shared__ float sC[3072];                    // A2f (2048) + Y2 (1024)
    __shared__ float mF[ZD], sf[ZD], b1f[ZD], c1[ZD], b2f[XD], c2[XD];
    __shared__ float red;
    const int tid = threadIdx.x;
    const float* A1t = ws + O_A1 + (size_t)t * ZD2;

    for (int i = tid; i < ZD2; i += 256) sA[i] = ws[O_PF + (size_t)t * ZD2 + i];
    if (tid < ZD) {
        mF[tid] = ws[O_FM + (size_t)(TSTEPS - 1) * ZD + tid];
        sf[tid] = ws[O_SF + (size_t)t * ZD + tid];
    }
    __syncthreads();
    mm_wmma(A1t, ZD, 0, sA, ZD, 0, sB, ZD, ZD, ZD, ZD);     // A1f
    mm_wmma(pC, ZD, 0, sA, ZD, 0, sC, ZD, XD, ZD, ZD);      // A2f (32x64)
    if (tid < ZD) {
        float acc = ws[O_B1 + (size_t)t * ZD + tid];
        for (int k = 0; k < ZD; ++k) acc += A1t[tid * ZD + k] * sf[k];
        b1f[tid] = acc;
    }
    if (tid >= 64 && tid < 96) {
        int j = tid - 64;
        float acc = pd[j] - obs[(size_t)t * XD + j];
        for (int k = 0; k < ZD; ++k) acc += pC[j * ZD + k] * sf[k];
        b2f[j] = acc;
    }
    __syncthreads();
    if (tid < ZD) {
        float acc = b1f[tid];
        for (int k = 0; k < ZD; ++k) acc += sB[tid * ZD + k] * mF[k];
        c1[tid] = acc;
    }
    if (tid >= 64 && tid < 96) {
        int j = tid - 64;
        float acc = b2f[j];
        for (int k = 0; k < ZD; ++k) acc += sC[j * ZD + k] * mF[k];
        c2[j] = acc;
    }
    __syncthreads();
    for (int i = tid; i < ZD2; i += 256) sA[i] = ws[O_FP + (size_t)(TSTEPS - 1) * ZD2 + i];
    __syncthreads();
    // t2 chain: X2 = A2f PF ; Y2 = X2 A2f^T
    mm_wmma(sC, ZD, 0, sA, ZD, 0, sD, ZD, XD, ZD, ZD);
    mm_wmma(sD, ZD, 0, sC, ZD, 1, sC + 2048, XD, XD, XD, ZD);
    // t1 chain (faithful to reference: NO transpose on trailing A1f)
    mm_wmma(sB, ZD, 0, sA, ZD, 0, sD, ZD, ZD, ZD, ZD);      // X = A1f PF
    mm_wmma(sD, ZD, 0, sB, ZD, 0, sA, ZD, ZD, ZD, ZD);      // Y = X A1f  (overwrites PF)
    if (tid == 0) red = 0.f;
    __syncthreads();
    float part = 0.f;
    const float* Prec1 = (t == 0) ? (ws + O_P0I) : (ws + O_PTR);
    for (int i = tid; i < ZD2; i += 256) {
        int a = i >> 6, b = i & 63;
        const float o1 = -0.5f * Prec1[a * ZD + b];
        part += o1 * sA[b * ZD + a];          // t1
        part += o1 * c1[a] * c1[b];           // v1
    }
    for (int i = tid; i < XD2; i += 256) {
        int a = i >> 5, b = i & 31;
        const float o2 = -0.5f * ws[O_PER + a * XD + b];
        part += o2 * sC[2048 + b * XD + a];   // t2
        part += o2 * c2[a] * c2[b];           // v2
    }
    atomicAdd(&red, part);
    __syncthreads();
    if (tid == 0) atomicAdd(&ws[O_SC + 4], red);
}

// ================= kernel 7: scalar assembly =================
__global__ __launch_bounds__(256)
void k_reduce(float* ws, float* out)
{
    __shared__ float aug[ZD * 128];
    __shared__ float ld, redB, redT;
    const int tid = threadIdx.x;
    for (int i = tid; i < ZD2; i += 256) {
        int r = i >> 6, c = i & 63;
        aug[r * 128 + c] = ws[O_FP + (size_t)(TSTEPS - 1) * ZD2 + i];
        aug[r * 128 + 64 + c] = (r == c) ? 1.f : 0.f;
    }
    if (tid == 0) { ld = 0.f; redB = 0.f; redT = 0.f; }
    __syncthreads();
    gauss_jordan(aug, ZD, &ld);                              // logdet(P_last)
    float pbv = 0.f, ptv = 0.f;
    for (int i = tid; i < TSTEPS - 1; i += 256) { pbv += ws[O_BLD + i]; ptv += ws[O_TRT + i]; }
    atomicAdd(&redB, pbv);
    atomicAdd(&redT, ptv);
    __syncthreads();
    if (tid == 0) {
        const float pTld = ws[O_SC + 0];
        const float pEld = ws[O_SC + 1];
        const float p0ld = ws[O_SC + 2];
        const float n1 = (float)(TSTEPS - 1);
        float consts = -0.5f * (ZD * F_LOG2PI + p0ld)
                     + -0.5f * (XD * F_LOG2PI + pEld)
                     + n1 * (-0.5f * (XD * F_LOG2PI + pEld)
                             - 0.5f * (ZD * F_LOG2PI + pTld)
                             + 0.5f * ZD)
                     + 0.5f * (n1 * ZD * F_LOG2PI + redB)
                     - 0.5f * redT
                     + ws[O_SC + 3];
        out[0] = consts
               + 0.5f * (ZD * F_LOG2PI + ld)
               + ws[O_SC + 4]
               + 0.5f * ZD;
    }
}

// ================= host entry =================
extern "C" void kernel_launch(void* const* d_in, const int* in_sizes, int n_in,
                              void* d_out, int out_size, void* d_ws, size_t ws_size,
                              hipStream_t stream)
{
    (void)in_sizes; (void)n_in; (void)out_size; (void)ws_size;
    const float* obs  = (const float*)d_in[0];
    const float* pA   = (const float*)d_in[1];
    const float* pb   = (const float*)d_in[2];
    const float* pQ   = (const float*)d_in[3];
    const float* pC   = (const float*)d_in[4];
    const float* pd   = (const float*)d_in[5];
    const float* pR   = (const float*)d_in[6];
    const float* pmu0 = (const float*)d_in[7];
    const float* pP0  = (const float*)d_in[8];
    const float* qA   = (const float*)d_in[9];
    const float* qb   = (const float*)d_in[10];
    const float* qQ   = (const float*)d_in[11];
    const float* qC   = (const float*)d_in[12];
    const float* qd   = (const float*)d_in[13];
    const float* qR   = (const float*)d_in[14];
    const float* qmu0 = (const float*)d_in[15];
    const float* qP0  = (const float*)d_in[16];
    float* ws  = (float*)d_ws;
    float* out = (float*)d_out;

    k_precompute<<<1, 256, 0, stream>>>(pA, pQ, pC, pR, pP0, qA, qQ, ws);
    k_kalman<<<1, 256, 0, stream>>>(obs, qA, qb, qQ, qC, qd, qR, qmu0, qP0, ws);
    k_bwd<<<TSTEPS - 1, 256, 0, stream>>>(qb, ws);
    k_prep<<<TSTEPS, 256, 0, stream>>>(pA, pb, pmu0, ws);
    k_scan2<<<2, 256, 0, stream>>>(ws);
    k_final<<<TSTEPS, 256, 0, stream>>>(obs, pC, pd, ws);
    k_reduce<<<1, 256, 0, stream>>>(ws, out);
}